// GNN_43130061586788
// MI455X (gfx1250) — compile-verified
//
#include <hip/hip_runtime.h>

typedef float v2f __attribute__((ext_vector_type(2)));
typedef float v8f __attribute__((ext_vector_type(8)));

#define LDS_ROW 132   // 128 + 4 pad -> bank-conflict-free column reads

// ---------------------------------------------------------------- fill ----
__global__ void gcn_fill_kernel(float* __restrict__ p, float v, long n) {
    long i = (long)blockIdx.x * blockDim.x + threadIdx.x;
    if (i < n) p[i] = v;
}

// ------------------------------------------------------------- degrees ----
__global__ void gcn_degree_kernel(const int* __restrict__ dst,
                                  const float* __restrict__ ew,
                                  float* __restrict__ deg, int E) {
    int e = blockIdx.x * blockDim.x + threadIdx.x;
    if (e < E) atomicAdd(&deg[dst[e]], ew[e]);
}

__global__ void gcn_rsqrt_kernel(float* __restrict__ deg, int n) {
    int i = blockIdx.x * blockDim.x + threadIdx.x;
    if (i < n) deg[i] = rsqrtf(deg[i]);   // deg >= 1.0 (self-loop), always > 0
}

// ------------------------------------------------- GEMM via f32 WMMA ------
// H[N x FOUT] = X[N x FIN] @ W[FIN x FOUT], fp32 exact (V_WMMA_F32_16X16X4_F32).
// Block = (FOUT/16) waves; wave w owns N-columns [16w, 16w+16).
// All FIN/4 B-fragments are hoisted into registers once per block and reused
// across MT M-tiles; per tile only the 16xFIN x-slab is staged in LDS.
template <int FIN, int MT>
__global__ void gcn_gemm_wmma_kernel(const float* __restrict__ X,
                                     const float* __restrict__ W,
                                     float* __restrict__ H,
                                     int N, int FOUT) {
    __shared__ float lds[16 * LDS_ROW];
    const int tid  = threadIdx.x;
    const int lane = tid & 31;
    const int wave = tid >> 5;

    // WMMA fragment mapping (ISA 7.12.2, 32-bit A 16x4 / B 4x16):
    //   lanes 0-15: K = k0+0, k0+1 ; lanes 16-31: K = k0+2, k0+3
    const int m    = lane & 15;
    const int koff = (lane < 16) ? 0 : 2;
    const int n    = wave * 16 + (lane & 15);

    // hoist all B fragments (identical for every M-tile): FIN/4 v2f in VGPRs
    v2f Bf[FIN / 4];
#pragma unroll
    for (int k = 0; k < FIN / 4; ++k) {
        Bf[k].x = W[(long)(4 * k + koff) * FOUT + n];
        Bf[k].y = W[(long)(4 * k + koff + 1) * FOUT + n];
    }

    const int rowoff = (lane < 16) ? 0 : 8;   // C/D: VGPR r -> M = r / r+8

    for (int t = 0; t < MT; ++t) {
        const int mbase = (blockIdx.x * MT + t) * 16;   // block-uniform
        if (mbase >= N) break;

        __syncthreads();   // protect LDS reuse from previous tile's readers
        for (int idx = tid; idx < 16 * FIN; idx += blockDim.x) {
            int r = idx / FIN, k = idx - r * FIN;
            int row = mbase + r;
            lds[r * LDS_ROW + k] = (row < N) ? X[(long)row * FIN + k] : 0.0f;
        }
        __syncthreads();

        v8f acc = {};
#pragma unroll
        for (int k = 0; k < FIN / 4; ++k) {
            v2f a;
            a.x = lds[m * LDS_ROW + 4 * k + koff];
            a.y = lds[m * LDS_ROW + 4 * k + koff + 1];
            // 8-arg form: (neg_a, A, neg_b, B, c_mod, C, reuse_a, reuse_b)
            acc = __builtin_amdgcn_wmma_f32_16x16x4_f32(
                false, a, false, Bf[k], (short)0, acc, false, false);
        }

#pragma unroll
        for (int r = 0; r < 8; ++r) {
            int row = mbase + r + rowoff;
            if (row < N) H[(long)row * FOUT + n] = acc[r];
        }
    }
}

// ------------------------------------------- edge gather/scale/scatter ----
// One wave per edge; F/32 floats per lane; atomic f32 adds resolved in L2
// (h and agg are ~51 MB each -> resident in the 192 MB global L2).
template <int F>
__global__ void gcn_scatter_kernel(const float* __restrict__ h,
                                   const int* __restrict__ src,
                                   const int* __restrict__ dst,
                                   const float* __restrict__ ew,
                                   const float* __restrict__ dinv,
                                   float* agg, int E) {
    constexpr int VEC = F / 32;
    const int wavesPerBlock = blockDim.x >> 5;
    const int e = blockIdx.x * wavesPerBlock + (threadIdx.x >> 5);
    if (e >= E) return;                   // wave-uniform exit
    const int lane = threadIdx.x & 31;

    const int s = src[e], d = dst[e];
    const float norm = dinv[s] * ew[e] * dinv[d];

    const float* hp = h   + (long)s * F + lane * VEC;
    float*       ap = agg + (long)d * F + lane * VEC;
#pragma unroll
    for (int j = 0; j < VEC; ++j)
        atomicAdd(&ap[j], norm * hp[j]);
}

// -------------------------------- self-loop + bias + ReLU (in place ok) ---
__global__ void gcn_epilogue_kernel(const float* agg,
                                    const float* __restrict__ h,
                                    const float* __restrict__ dinv,
                                    const float* __restrict__ b,
                                    float* y, int F) {
    const int i = blockIdx.x;        // node
    const int f = threadIdx.x;       // feature (blockDim == F)
    const long id = (long)i * F + f;
    const float di = dinv[i];
    const float v = agg[id] + di * di * h[id] + b[f];
    y[id] = v > 0.0f ? v : 0.0f;
}

// -------------------------------------------------------------- launch ----
extern "C" void kernel_launch(void* const* d_in, const int* in_sizes, int n_in,
                              void* d_out, int out_size, void* d_ws, size_t ws_size,
                              hipStream_t stream) {
    const float* x  = (const float*)d_in[0];
    const int*   ei = (const int*)d_in[1];     // [2, E] int32
    const float* ew = (const float*)d_in[2];
    const float* W1 = (const float*)d_in[3];
    const float* b1 = (const float*)d_in[4];
    const float* W2 = (const float*)d_in[5];
    const float* b2 = (const float*)d_in[6];
    float* out = (float*)d_out;

    const int F1  = in_sizes[4];               // 128
    const int F2  = in_sizes[6];               // 64
    const int FIN = in_sizes[3] / F1;          // 128
    const int N   = in_sizes[0] / FIN;         // 100000
    const int E   = in_sizes[2];               // 600000
    const int* src = ei;
    const int* dst = ei + E;

    // workspace carve-up (fp32): deg | h1 | agg1(y1) | h2  (~128.5 MB)
    float* ws   = (float*)d_ws;
    float* deg  = ws;
    float* h1   = ws + (((long)N + 255) & ~255L);
    float* agg1 = h1 + (long)N * F1;
    float* h2   = agg1 + (long)N * F1;

    const int T = 256;
    const long n1 = (long)N * F1, n2 = (long)N * F2;

    // init: deg = 1.0 (self-loop weight folded in), accumulators zeroed
    gcn_fill_kernel<<<(int)((N + T - 1) / T), T, 0, stream>>>(deg, 1.0f, N);
    gcn_fill_kernel<<<(int)((n1 + T - 1) / T), T, 0, stream>>>(agg1, 0.0f, n1);
    gcn_fill_kernel<<<(int)((n2 + T - 1) / T), T, 0, stream>>>(out, 0.0f, n2);

    // symmetric degree normalization
    gcn_degree_kernel<<<(E + T - 1) / T, T, 0, stream>>>(dst, ew, deg, E);
    gcn_rsqrt_kernel<<<(N + T - 1) / T, T, 0, stream>>>(deg, N);

    constexpr int MT = 8;                      // M-tiles per block
    const int mtiles = (N + 15) / 16;
    const int gblocks = (mtiles + MT - 1) / MT;

    // ---- layer 1: h1 = x @ W1 ; aggregate ; relu(+b1) -> agg1 (in place)
    gcn_gemm_wmma_kernel<128, MT><<<gblocks, (F1 / 16) * 32, 0, stream>>>(x, W1, h1, N, F1);
    gcn_scatter_kernel<128><<<(E + 7) / 8, 256, 0, stream>>>(h1, src, dst, ew, deg, agg1, E);
    gcn_epilogue_kernel<<<N, F1, 0, stream>>>(agg1, h1, deg, b1, agg1, F1);

    // ---- layer 2: h2 = y1 @ W2 ; aggregate into d_out ; relu(+b2) in place
    gcn_gemm_wmma_kernel<128, MT><<<gblocks, (F2 / 16) * 32, 0, stream>>>(agg1, W2, h2, N, F1);
    gcn_scatter_kernel<64><<<(E + 7) / 8, 256, 0, stream>>>(h2, src, dst, ew, deg, out, E);
    gcn_epilogue_kernel<<<N, F2, 0, stream>>>(out, h2, deg, b2, out, F2);
}